// TopFeatsSelector_10471130268337
// MI455X (gfx1250) — compile-verified
//
#include <hip/hip_runtime.h>
#include <math.h>

#define B_ 32
#define H_ 16
#define S_ 577
#define D_ 1024
#define TOPK 289
#define NPATCH (S_ - 1) /* 576 */

typedef __attribute__((ext_vector_type(2))) float v2f;
typedef __attribute__((ext_vector_type(8))) float v8f;
typedef __attribute__((ext_vector_type(4))) unsigned int v4u;
typedef __attribute__((ext_vector_type(8))) int v8i_t;
typedef __attribute__((ext_vector_type(4))) int v4i_t;

// ---------------------------------------------------------------------------
// Kernel 1: per batch, compute maws[b, i] (i = 0..575), ordering-equivalent to
//   softmax_q(mean_h pre[b,:,q,0])[i+1] * mean_h post[b,h,0,i+1]
// Head reduction of the cls COLUMN done with V_WMMA_F32_16X16X4_F32:
//   C(16q x 16) = sum over 4 k-chunks of A(16q x 4heads) * ones(4 x 16)
// ---------------------------------------------------------------------------
__global__ __launch_bounds__(640) void maws_kernel(
    const float* __restrict__ pre, const float* __restrict__ post,
    float* __restrict__ maws) {
  __shared__ float col[592];  // 37 tiles * 16 = 592; only [0..576] is ever read
  __shared__ float red[512];

  const int b = blockIdx.x;
  const int tid = threadIdx.x;
  const int wave = tid >> 5;
  const int lane = tid & 31;
  const size_t SS = (size_t)S_ * (size_t)S_;
  const float* preB = pre + (size_t)b * H_ * SS;

  // A-matrix 16x4 f32 layout: lanes 0-15 hold K=0,1 ; lanes 16-31 hold K=2,3
  const int m = lane & 15;
  const int koff = (lane < 16) ? 0 : 2;
  v2f ones;
  ones.x = 1.0f;
  ones.y = 1.0f;

  // 37 tiles of 16 q's cover q = 0..576 (tile 36 clamped)
  for (int t = wave; t < 37; t += 20) {
    int q = t * 16 + m;
    if (q > 576) q = 576;  // clamp; clamped lanes land in col[577..591], never read
    const float* pq = preB + (size_t)q * S_;  // column element (q, 0); head h at +h*SS
    v8f acc = {};
#pragma unroll
    for (int c = 0; c < 4; ++c) {
      const int h = c * 4 + koff;
      v2f a;
      a.x = pq[(size_t)h * SS];
      a.y = pq[(size_t)(h + 1) * SS];
      acc = __builtin_amdgcn_wmma_f32_16x16x4_f32(
          /*neg_a=*/false, a, /*neg_b=*/false, ones,
          /*c_mod=*/(short)0, acc, /*reuse_a=*/false, /*reuse_b=*/false);
    }
    // C layout: VGPR r, lanes 0-15 -> (M=r, N=lane); lanes 16-31 -> (M=r+8).
    // Column N=0 lives in lane 0 (M=0..7) and lane 16 (M=8..15).
    if ((lane & 15) == 0) {
      const int base = t * 16 + ((lane >> 4) << 3);  // lane0 -> +0, lane16 -> +8
#pragma unroll
      for (int r = 0; r < 8; ++r) col[base + r] = acc[r];
    }
  }
  __syncthreads();

  // block max over col[0..576]
  if (tid < 512) {
    float v = col[tid];
    if (tid + 512 < S_) v = fmaxf(v, col[tid + 512]);
    red[tid] = v;
  }
  __syncthreads();
  for (int s = 256; s > 0; s >>= 1) {
    if (tid < s) red[tid] = fmaxf(red[tid], red[tid + s]);
    __syncthreads();
  }
  const float cmax = red[0] * (1.0f / 16.0f);

  if (tid < NPATCH) {
    const int i = tid;
    // coalesced head-sum of the cls ROW of post_sm_attn
    const float* pr = post + (size_t)b * H_ * SS + (size_t)(i + 1);
    float ps = 0.0f;
#pragma unroll
    for (int h = 0; h < H_; ++h) ps += pr[(size_t)h * SS];
    const float cm = col[i + 1] * (1.0f / 16.0f);  // mean over heads
    // drop 1/Z (global positive scale); keep exp(mean - max) * sum_post
    maws[(size_t)b * NPATCH + i] = __expf(cm - cmax) * ps;
  }
}

// ---------------------------------------------------------------------------
// Kernel 2: stable top-k ranking (ties -> lower index first, as lax.top_k).
// Candidate i goes to output slot rank(i) if rank < TOPK.
// ---------------------------------------------------------------------------
__global__ __launch_bounds__(NPATCH) void rank_kernel(
    const float* __restrict__ maws, int* __restrict__ idx) {
  __shared__ float mv[NPATCH];
  const int b = blockIdx.x;
  const int tid = threadIdx.x;
  mv[tid] = maws[(size_t)b * NPATCH + tid];
  __syncthreads();
  const float v = mv[tid];
  int rank = 0;
  for (int j = 0; j < NPATCH; ++j) {
    const float u = mv[j];
    rank += (int)((u > v) || (u == v && j < tid));
  }
  if (rank < TOPK) idx[b * TOPK + rank] = tid;
}

// ---------------------------------------------------------------------------
// Kernel 3: gather feats rows via the Tensor Data Mover in GATHER mode.
// One block per (16-row chunk, batch). Wave 0 builds the D# (16-bit row
// indices, data_size=4B, tile_dim0=D), issues tensor_load_to_lds, waits on
// TENSORcnt, then the whole block streams LDS -> out with float4 stores.
// ---------------------------------------------------------------------------
__global__ __launch_bounds__(512) void gather_kernel(
    const float* __restrict__ feats, const int* __restrict__ idx,
    float* __restrict__ out) {
  __shared__ float tile[16 * D_];  // 64 KB

  const int chunk = blockIdx.x;
  const int b = blockIdx.y;
  const int r0 = chunk * 16;
  const int n = min(16, TOPK - r0);
  const int tid = threadIdx.x;

  if (tid < 32) {  // single wave issues the TDM op
    unsigned rowidx[16];
#pragma unroll
    for (int k = 0; k < 16; ++k)
      rowidx[k] = (k < n) ? (unsigned)idx[b * TOPK + r0 + k] : 0u;

    const unsigned long long ga =
        (unsigned long long)(const void*)(feats + (size_t)b * S_ * D_);
    const unsigned lds_base = (unsigned)(size_t)(&tile[0]);

    // D# group 0: count=1, gather_mode=1, 16-bit indices, lds addr, global addr
    v4u g0;
    g0.x = 0x80000001u;
    g0.y = lds_base;
    g0.z = (unsigned)ga;
    g0.w = (unsigned)((ga >> 32) & 0x1FFFFFFull) | (2u << 30);  // type=2

    // D# group 1
    v8i_t g1;
    g1.s0 = (int)(2u << 16);                         // data_size = 4 bytes
    g1.s1 = (int)(((unsigned)D_ & 0xFFFFu) << 16);   // tensor_dim0[15:0]
    g1.s2 = (int)((((unsigned)D_ >> 16) & 0xFFFFu) |
                  (((unsigned)NPATCH & 0xFFFFu) << 16));  // dim0 hi, dim1 lo
    g1.s3 = (int)((((unsigned)NPATCH >> 16) & 0xFFFFu) |
                  (((unsigned)D_) << 16));           // dim1 hi, tile_dim0 = D
    g1.s4 = (int)(unsigned)n;                        // tile_dim1 = #valid indices
    g1.s5 = (int)(unsigned)D_;                       // tensor_dim0_stride lo32
    g1.s6 = 0;
    g1.s7 = 0;

    // D# groups 2-3: sixteen 16-bit row indices
    v4i_t g2, g3;
    g2.x = (int)(rowidx[0] | (rowidx[1] << 16));
    g2.y = (int)(rowidx[2] | (rowidx[3] << 16));
    g2.z = (int)(rowidx[4] | (rowidx[5] << 16));
    g2.w = (int)(rowidx[6] | (rowidx[7] << 16));
    g3.x = (int)(rowidx[8] | (rowidx[9] << 16));
    g3.y = (int)(rowidx[10] | (rowidx[11] << 16));
    g3.z = (int)(rowidx[12] | (rowidx[13] << 16));
    g3.w = (int)(rowidx[14] | (rowidx[15] << 16));

    // 6-arg form (clang-23 / therock headers): extra int32x8 group before cpol
    v8i_t g4 = {};
    __builtin_amdgcn_tensor_load_to_lds(g0, g1, g2, g3, g4, /*cpol=*/0);
    __builtin_amdgcn_s_wait_tensorcnt(0);
  }
  __syncthreads();

  // stream gathered rows out (contiguous n*4KB region)
  const int n4 = n * (D_ / 4);
  const float4* __restrict__ src = (const float4*)tile;
  float4* __restrict__ dst = (float4*)(out + ((size_t)b * TOPK + r0) * D_);
  for (int j = tid; j < n4; j += 512) dst[j] = src[j];
}

// ---------------------------------------------------------------------------
extern "C" void kernel_launch(void* const* d_in, const int* in_sizes, int n_in,
                              void* d_out, int out_size, void* d_ws,
                              size_t ws_size, hipStream_t stream) {
  const float* feats = (const float*)d_in[0];        // [B,S,D]
  const float* pre = (const float*)d_in[1];          // [B,H,S,S]
  const float* post = (const float*)d_in[2];         // [B,H,S,S]
  float* out = (float*)d_out;                        // [B,TOPK,D]

  float* maws = (float*)d_ws;                                          // B*576 f32
  int* idx = (int*)((char*)d_ws + (size_t)B_ * NPATCH * sizeof(float)); // B*289 i32

  maws_kernel<<<B_, 640, 0, stream>>>(pre, post, maws);
  rank_kernel<<<B_, NPATCH, 0, stream>>>(maws, idx);
  gather_kernel<<<dim3((TOPK + 15) / 16, B_), 512, 0, stream>>>(feats, idx, out);
}